// FernSparseTable_44779329028743
// MI455X (gfx1250) — compile-verified
//
#include <hip/hip_runtime.h>

#ifndef __has_builtin
#define __has_builtin(x) 0
#endif

#define NIMG  32
#define MFERN 16
#define KBITS 10
#define LPICK 3
#define PCOMB 8
#define DOUT  64
#define TAB   1024
#define HW    4096
#define ROWDW 68                        // padded row pitch in dwords (272B, 16B aligned)
#define LDS_BYTES (TAB * ROWDW * 4)     // 278528 B of the 320KB WGP LDS

extern __shared__ float smem[];

typedef int v4i __attribute__((vector_size(16)));
typedef __attribute__((address_space(1))) v4i* gv4i_p;   // global ptr to int4
typedef __attribute__((address_space(3))) v4i* lv4i_p;   // LDS ptr to int4

__device__ __forceinline__ void stage_chunk_b128(const float* gsrc, unsigned lds_byte) {
#if __has_builtin(__builtin_amdgcn_global_load_async_to_lds_b128)
    __builtin_amdgcn_global_load_async_to_lds_b128(
        (gv4i_p)(gsrc),
        (lv4i_p)(((char*)smem) + lds_byte),
        0, 0);
#else
    asm volatile("global_load_async_to_lds_b128 %0, %1, off"
                 :
                 : "v"(lds_byte), "v"(gsrc)
                 : "memory");
#endif
}

__device__ __forceinline__ void wait_async0() {
#if __has_builtin(__builtin_amdgcn_s_wait_asynccnt)
    __builtin_amdgcn_s_wait_asynccnt(0);
#else
    asm volatile("s_wait_asynccnt 0x0" ::: "memory");
#endif
}

__global__ __launch_bounds__(512)
void fern_sparse_table_kernel(const float* __restrict__ B,
                              const float* __restrict__ W,
                              const float* __restrict__ bias,
                              float* __restrict__ out) {
    const int tid = threadIdx.x;                 // 0..511
    const int n   = blockIdx.x >> 3;             // image
    const int hw  = ((blockIdx.x & 7) << 9) + tid;

    float acc[DOUT];
#pragma unroll
    for (int d = 0; d < DOUT; ++d) acc[d] = 0.f;

#pragma unroll 1
    for (int m = 0; m < MFERN; ++m) {
        if (m) __syncthreads();                  // prior fern's table no longer in use

        // ---- async DMA stage W[m] (1024x64 f32) into LDS, 4-dword row pad ----
        const float* Wm = W + (size_t)m * TAB * DOUT;
#pragma unroll 4
        for (int t = 0; t < 32; ++t) {
            int c   = tid + (t << 9);            // 16-byte chunk id, 0..16383
            int row = c >> 4;
            int col = (c & 15) << 2;             // dword column 0..60
            stage_chunk_b128(Wm + row * DOUT + col,
                             (unsigned)((row * ROWDW + col) << 2));
        }
        wait_async0();
        __syncthreads();

        // ---- per-pixel fern evaluation (one pixel per lane) ----
        const float* Bp = B + ((size_t)n * (MFERN * KBITS) + (size_t)m * KBITS) * HW + hw;
        if (m + 1 < MFERN) __builtin_prefetch(Bp + (size_t)KBITS * HW, 0, 0);

        float T[KBITS], ba[KBITS];
        int   wb  = 0;
        float bsp = 1.f;
#pragma unroll
        for (int k = 0; k < KBITS; ++k) {
            float tk = __builtin_nontemporal_load(Bp + (size_t)k * HW);
            T[k]  = tk;
            wb   |= (tk >= 0.5f ? 1 : 0) << k;
            bsp  *= fmaxf(tk, 1.f - tk);
            ba[k] = fabsf(tk - 0.5f);
        }

        int abi[LPICK]; float aba[LPICK];
#pragma unroll
        for (int j = 0; j < LPICK; ++j) {
            float best = ba[0]; float bt = T[0]; int bi = 0;
#pragma unroll
            for (int k = 1; k < KBITS; ++k) {
                bool c = ba[k] < best;           // strict < keeps first-index on ties
                best = c ? ba[k] : best;
                bt   = c ? T[k]  : bt;
                bi   = c ? k     : bi;
            }
            abi[j] = bi; aba[j] = bt;
#pragma unroll
            for (int k = 0; k < KBITS; ++k) ba[k] += (k == bi) ? 1.f : 0.f;
            bsp /= fmaxf(bt, 1.f - bt);          // renormalize bsp
        }

#pragma unroll
        for (int p = 0; p < PCOMB; ++p) {
            float at = bsp;
            int   it = wb;
#pragma unroll
            for (int j = 0; j < LPICK; ++j) {
                bool bit = (p >> j) & 1;
                at *= bit ? aba[j] : (1.f - aba[j]);
                int msk = 1 << abi[j];
                it = bit ? (it | msk) : (it & ~msk);
            }
            const float4* rowp = reinterpret_cast<const float4*>(smem + it * ROWDW);
#pragma unroll
            for (int c = 0; c < 16; ++c) {
                float4 v = rowp[c];
                acc[4 * c + 0] = fmaf(at, v.x, acc[4 * c + 0]);
                acc[4 * c + 1] = fmaf(at, v.y, acc[4 * c + 1]);
                acc[4 * c + 2] = fmaf(at, v.z, acc[4 * c + 2]);
                acc[4 * c + 3] = fmaf(at, v.w, acc[4 * c + 3]);
            }
        }
    }

    // ---- write out[n, d, h, w] once, non-temporal (streaming) ----
    float* op = out + (size_t)n * DOUT * HW + hw;
#pragma unroll
    for (int d = 0; d < DOUT; ++d) {
        __builtin_nontemporal_store(acc[d] + bias[d], op + (size_t)d * HW);
    }
}

extern "C" void kernel_launch(void* const* d_in, const int* in_sizes, int n_in,
                              void* d_out, int out_size, void* d_ws, size_t ws_size,
                              hipStream_t stream) {
    const float* B    = (const float*)d_in[0];  // (32, 160, 64, 64)
    const float* W    = (const float*)d_in[1];  // (16, 1024, 64)
    const float* bias = (const float*)d_in[2];  // (64,)
    float*       out  = (float*)d_out;          // (32, 64, 64, 64)

    dim3 grid(NIMG * 8);     // 256 blocks: 8 hw-tiles of 512 px per image
    dim3 block(512);
    hipLaunchKernelGGL(fern_sparse_table_kernel, grid, block, LDS_BYTES, stream,
                       B, W, bias, out);
}